// WassersteinExp_34591666602826
// MI455X (gfx1250) — compile-verified
//
#include <hip/hip_runtime.h>
#include <hip/hip_bf16.h>
#include <stdint.h>

#define BLK 256

#if defined(__has_builtin)
#  if __has_builtin(__builtin_amdgcn_tensor_load_to_lds)
#    define HAVE_TDM 1
#  endif
#  if __has_builtin(__builtin_amdgcn_s_wait_tensorcnt)
#    define HAVE_TWAIT 1
#  endif
#  if __has_builtin(__builtin_amdgcn_rcpf)
#    define FAST_RCP(x) __builtin_amdgcn_rcpf(x)
#  endif
#  if __has_builtin(__builtin_amdgcn_cosf)
#    define HAVE_HW_COS 1
#  endif
#endif
#ifndef HAVE_TDM
#  define HAVE_TDM 0
#endif
#ifndef HAVE_TWAIT
#  define HAVE_TWAIT 0
#endif
#ifndef FAST_RCP
#  define FAST_RCP(x) (1.0f / (x))
#endif
#ifndef HAVE_HW_COS
#  define HAVE_HW_COS 0
#endif

typedef unsigned int u32x4 __attribute__((ext_vector_type(4)));
typedef int          i32x8 __attribute__((ext_vector_type(8)));
typedef int          i32x4 __attribute__((ext_vector_type(4)));

// cos with argument given in revolutions (x*2*pi radians); valid for |x| small.
__device__ __forceinline__ float cos_rev(float x_rev) {
#if HAVE_HW_COS
  return __builtin_amdgcn_cosf(x_rev);        // v_cos_f32: takes its input in revolutions
#else
  return cosf(x_rev * 6.2831853071795864f);
#endif
}

#if HAVE_TDM
// 1-D TDM transfer: copy `elems` consecutive f32 from global to LDS.
// D# layout per CDNA5 ISA. tile_dim0==0 => NOP.
__device__ __forceinline__ void tdm_load_f32(uint32_t lds_byte_addr,
                                             const float* gsrc,
                                             uint32_t elems) {
  uint64_t ga = (uint64_t)(uintptr_t)gsrc;
  u32x4 g0 = { 1u,                                   // count=1 (valid)
               lds_byte_addr,                        // bits 63:32 lds_addr
               (uint32_t)ga,                         // global_addr[31:0]
               (uint32_t)((ga >> 32) & 0x01FFFFFFu)  // global_addr[56:32]
               | 0x80000000u };                      // type=2 at bits 127:126
  i32x8 g1 = { (int)0x00020000u,                     // data_size=4B
               (int)((elems & 0xFFFFu) << 16),       // tensor_dim0[15:0]
               (int)(((elems >> 16) & 0xFFFFu)
                     | (1u << 16)),                  // tensor_dim0 hi | tensor_dim1=1
               (int)((elems & 0xFFFFu) << 16),       // tile_dim0 = elems
               0,                                    // tile_dim1/2 = 0
               (int)elems,                           // tensor_dim0_stride
               0, 0 };
  i32x4 z4 = {0, 0, 0, 0};
#if defined(__clang_major__) && (__clang_major__ >= 23)
  i32x8 z8 = {0, 0, 0, 0, 0, 0, 0, 0};
  __builtin_amdgcn_tensor_load_to_lds(g0, g1, z4, z4, z8, 0);
#else
  __builtin_amdgcn_tensor_load_to_lds(g0, g1, z4, z4, 0);
#endif
}
#endif

__device__ __forceinline__ void wait_tensorcnt0() {
#if HAVE_TWAIT
  __builtin_amdgcn_s_wait_tensorcnt(0);
#elif HAVE_TDM
  asm volatile("s_wait_tensorcnt 0" ::: "memory");
#endif
}

// Eigenvector of symmetric 3x3 for eigenvalue lam (cross-product of rows of A-lam*I).
__device__ __forceinline__ void eig_vec(float d0, float d1, float d2,
                                        float s01, float s02, float s12,
                                        float lam, float v[3]) {
  float r0x = d0 - lam, r0y = s01,      r0z = s02;
  float r1x = s01,      r1y = d1 - lam, r1z = s12;
  float r2x = s02,      r2y = s12,      r2z = d2 - lam;
  float cx0 = r0y*r1z - r0z*r1y, cy0 = r0z*r1x - r0x*r1z, cz0 = r0x*r1y - r0y*r1x;
  float cx1 = r0y*r2z - r0z*r2y, cy1 = r0z*r2x - r0x*r2z, cz1 = r0x*r2y - r0y*r2x;
  float cx2 = r1y*r2z - r1z*r2y, cy2 = r1z*r2x - r1x*r2z, cz2 = r1x*r2y - r1y*r2x;
  float n0 = cx0*cx0 + cy0*cy0 + cz0*cz0;
  float n1 = cx1*cx1 + cy1*cy1 + cz1*cz1;
  float n2 = cx2*cx2 + cy2*cy2 + cz2*cz2;
  float bx = cx0, by = cy0, bz = cz0, bn = n0;
  if (n1 > bn) { bx = cx1; by = cy1; bz = cz1; bn = n1; }
  if (n2 > bn) { bx = cx2; by = cy2; bz = cz2; bn = n2; }
  if (bn > 1e-30f) {
    float inv = rsqrtf(bn);
    v[0] = bx * inv; v[1] = by * inv; v[2] = bz * inv;
  } else {
    v[0] = 1.f; v[1] = 0.f; v[2] = 0.f;   // (near-)isotropic fallback
  }
}

__global__ __launch_bounds__(BLK) void wexp_kernel(
    const float* __restrict__ g_loc,  const float* __restrict__ g_cov,
    const float* __restrict__ g_vel,  const float* __restrict__ g_vcov,
    float* __restrict__ o_loc, float* __restrict__ o_cov, int n) {
  __shared__ float sA[BLK * 9];
  __shared__ float sV[BLK * 9];

  const int tid  = (int)threadIdx.x;
  const int base = (int)blockIdx.x * BLK;
  int remi = n - base; if (remi < 0) remi = 0; if (remi > BLK) remi = BLK;
  const uint32_t rem_rows = (uint32_t)remi;

#if HAVE_TDM
  {
    // Each of the 8 wave32s DMAs its own 32-row (288-float) slice of both arrays.
    const uint32_t wave  = (uint32_t)tid >> 5;
    const uint32_t wrow0 = wave * 32u;
    uint32_t wr = (rem_rows > wrow0) ? (rem_rows - wrow0) : 0u;
    if (wr > 32u) wr = 32u;
    const uint32_t v    = wr * 9u;           // elements this wave moves (0 => NOP)
    const uint32_t woff = wrow0 * 9u;        // element offset within block tile
    const uint32_t ldsA = (uint32_t)(uintptr_t)(void*)sA + woff * 4u;
    const uint32_t ldsV = (uint32_t)(uintptr_t)(void*)sV + woff * 4u;
    tdm_load_f32(ldsA, g_cov  + (size_t)base * 9 + woff, v);
    tdm_load_f32(ldsV, g_vcov + (size_t)base * 9 + woff, v);
    wait_tensorcnt0();
  }
  __syncthreads();
#else
  for (uint32_t i = (uint32_t)tid; i < rem_rows * 9u; i += BLK) {
    sA[i] = g_cov[(size_t)base * 9 + i];
    sV[i] = g_vcov[(size_t)base * 9 + i];
  }
  __syncthreads();
#endif

  const int row = base + tid;
  if (row >= n) return;

  // --- register loads (stride-9 LDS reads are bank-conflict-free: gcd(9,64)=1) ---
  float a[9], w[9];
#pragma unroll
  for (int i = 0; i < 9; i++) { a[i] = sA[tid * 9 + i]; w[i] = sV[tid * 9 + i]; }

  // --- new_loc = loc + velocity ---
#pragma unroll
  for (int c = 0; c < 3; c++)
    o_loc[(size_t)row * 3 + c] = g_loc[(size_t)row * 3 + c] + g_vel[(size_t)row * 3 + c];

  // --- stability mask on cov1 ---
  bool m = true;
  float amax = 0.f, amin = 3.4e38f;
#pragma unroll
  for (int i = 0; i < 9; i++) {
    float av = fabsf(a[i]);
    amax = fmaxf(amax, av); amin = fminf(amin, av);
    m = m && __builtin_isfinite(a[i]);
  }
  m = m && (amax < 1e4f) && (amin > 1e-8f);
  float asym = fmaxf(fabsf(a[1] - a[3]), fmaxf(fabsf(a[2] - a[6]), fabsf(a[5] - a[7])));
  m = m && (asym < 1e-5f);
  float det = a[0]*(a[4]*a[8] - a[5]*a[7]) - a[1]*(a[3]*a[8] - a[5]*a[6])
            + a[2]*(a[3]*a[7] - a[4]*a[6]);
  m = m && (det > 1e-8f);

  // --- analytic symmetric 3x3 eigendecomposition ---
  const float d0 = a[0], d1 = a[4], d2 = a[8];
  const float s01 = 0.5f*(a[1] + a[3]), s02 = 0.5f*(a[2] + a[6]), s12 = 0.5f*(a[5] + a[7]);
  const float q  = (d0 + d1 + d2) * (1.f / 3.f);
  const float b0 = d0 - q, b1 = d1 - q, b2 = d2 - q;
  const float p1 = s01*s01 + s02*s02 + s12*s12;
  const float p2 = b0*b0 + b1*b1 + b2*b2 + 2.f*p1;
  const float z  = fmaxf(p2, 1e-24f) * (1.f / 6.f);
  const float invp = rsqrtf(z);          // 1/p  (one v_rsq)
  const float p    = z * invp;           // p = sqrt(z)
  const float c00 = b0*invp, c11 = b1*invp, c22 = b2*invp;
  const float c01 = s01*invp, c02 = s02*invp, c12v = s12*invp;
  float r = 0.5f * (c00*(c11*c22 - c12v*c12v) - c01*(c01*c22 - c12v*c02)
                    + c02*(c01*c12v - c11*c02));
  r = fminf(1.f, fmaxf(-1.f, r));
  // phi in [0, pi/3] -> revolutions in [0, 1/6]: raw v_cos_f32 is safe, no range reduction.
  const float phi_rev = acosf(r) * (1.f / 3.f) * 0.15915494309189535f;
  const float e1 = q + 2.f*p*cos_rev(phi_rev);                  // largest
  const float e3 = q + 2.f*p*cos_rev(phi_rev + (1.f/3.f));      // smallest
  const float e2 = 3.f*q - e1 - e3;
  float lam[3] = { e3, e2, e1 };                                // ascending (eigh order)

  float v0[3], v2[3], v1[3];
  eig_vec(d0, d1, d2, s01, s02, s12, lam[0], v0);
  eig_vec(d0, d1, d2, s01, s02, s12, lam[2], v2);
  {
    float cxp = v2[1]*v0[2] - v2[2]*v0[1];
    float cyp = v2[2]*v0[0] - v2[0]*v0[2];
    float czp = v2[0]*v0[1] - v2[1]*v0[0];
    float nn  = cxp*cxp + cyp*cyp + czp*czp;
    if (nn > 1e-30f) {
      float inv = rsqrtf(nn);
      v1[0] = cxp*inv; v1[1] = cyp*inv; v1[2] = czp*inv;
    } else { v1[0] = 0.f; v1[1] = 1.f; v1[2] = 0.f; }
  }
  float R[3][3];   // columns = eigenvectors
#pragma unroll
  for (int i = 0; i < 3; i++) { R[i][0] = v0[i]; R[i][1] = v1[i]; R[i][2] = v2[i]; }

  // --- remaining mask terms ---
  float smax = fmaxf(fabsf(lam[0]), fmaxf(fabsf(lam[1]), fabsf(lam[2])));
  float smin = fminf(fabsf(lam[0]), fminf(fabsf(lam[1]), fabsf(lam[2])));
  m = m && (smax < 1e4f) && (smin > 1e-8f);
  float vmax = 0.f;
#pragma unroll
  for (int i = 0; i < 9; i++) {
    vmax = fmaxf(vmax, fabsf(w[i]));
    m = m && __builtin_isfinite(w[i]);
  }
  m = m && (vmax < 1e4f);
  float vsym = fmaxf(fabsf(w[1] - w[3]), fmaxf(fabsf(w[2] - w[6]), fabsf(w[5] - w[7])));
  m = m && (vsym < 1e-5f);
#pragma unroll
  for (int i = 0; i < 3; i++)
#pragma unroll
    for (int j = 0; j < 3; j++) m = m && __builtin_isfinite(R[i][j]);

  // --- Wasserstein exponential map (all intermediates are symmetric: compute 6 entries) ---
  float V[3][3] = { { w[0], w[1], w[2] }, { w[3], w[4], w[5] }, { w[6], w[7], w[8] } };

  float T1[3][3];                       // V * R
#pragma unroll
  for (int i = 0; i < 3; i++)
#pragma unroll
    for (int j = 0; j < 3; j++)
      T1[i][j] = V[i][0]*R[0][j] + V[i][1]*R[1][j] + V[i][2]*R[2][j];

  float C[3][3];                        // R^T V R  (symmetric)
#pragma unroll
  for (int i = 0; i < 3; i++)
#pragma unroll
    for (int j = 0; j < 3; j++)
      if (j >= i) {
        float c = R[0][i]*T1[0][j] + R[1][i]*T1[1][j] + R[2][i]*T1[2][j];
        C[i][j] = c; C[j][i] = c;
      }

  float E[3][3];                        // C / (lam_i + lam_j + eps)  (symmetric)
#pragma unroll
  for (int i = 0; i < 3; i++)
#pragma unroll
    for (int j = 0; j < 3; j++)
      if (j >= i) {
        float e = C[i][j] * FAST_RCP(lam[i] + lam[j] + 1e-8f);   // 6 v_rcp
        E[i][j] = e; E[j][i] = e;
      }

  float T2[3][3];                       // E * R^T
#pragma unroll
  for (int i = 0; i < 3; i++)
#pragma unroll
    for (int j = 0; j < 3; j++)
      T2[i][j] = E[i][0]*R[j][0] + E[i][1]*R[j][1] + E[i][2]*R[j][2];

  float G[3][3];                        // gamma = R E R^T  (symmetric)
#pragma unroll
  for (int i = 0; i < 3; i++)
#pragma unroll
    for (int j = 0; j < 3; j++)
      if (j >= i) {
        float g = R[i][0]*T2[0][j] + R[i][1]*T2[1][j] + R[i][2]*T2[2][j];
        G[i][j] = g; G[j][i] = g;
      }

  float Rl[3][3];
#pragma unroll
  for (int i = 0; i < 3; i++)
#pragma unroll
    for (int k = 0; k < 3; k++) Rl[i][k] = R[i][k] * lam[k];

  float S[3][3];                        // R diag(lam) R^T  (symmetric)
#pragma unroll
  for (int i = 0; i < 3; i++)
#pragma unroll
    for (int j = 0; j < 3; j++)
      if (j >= i) {
        float s = Rl[i][0]*R[j][0] + Rl[i][1]*R[j][1] + Rl[i][2]*R[j][2];
        S[i][j] = s; S[j][i] = s;
      }

  float U[3][3];                        // G * S
#pragma unroll
  for (int i = 0; i < 3; i++)
#pragma unroll
    for (int j = 0; j < 3; j++)
      U[i][j] = G[i][0]*S[0][j] + G[i][1]*S[1][j] + G[i][2]*S[2][j];

  float T[3][3];                        // G S G^T  (symmetric)
#pragma unroll
  for (int i = 0; i < 3; i++)
#pragma unroll
    for (int j = 0; j < 3; j++)
      if (j >= i) {
        float t = U[i][0]*G[j][0] + U[i][1]*G[j][1] + U[i][2]*G[j][2];
        T[i][j] = t; T[j][i] = t;
      }

  float upd[9];
#pragma unroll
  for (int i = 0; i < 3; i++)
#pragma unroll
    for (int j = 0; j < 3; j++)
      upd[i*3 + j] = S[i][j] + V[i][j] + T[i][j];

#pragma unroll
  for (int i = 0; i < 9; i++)
    o_cov[(size_t)row * 9 + i] = m ? upd[i] : a[i];
}

extern "C" void kernel_launch(void* const* d_in, const int* in_sizes, int n_in,
                              void* d_out, int out_size, void* d_ws, size_t ws_size,
                              hipStream_t stream) {
  const float* loc  = (const float*)d_in[0];
  const float* cov1 = (const float*)d_in[1];
  const float* vel  = (const float*)d_in[2];
  const float* vcov = (const float*)d_in[3];
  const int n = in_sizes[0] / 3;
  float* o_loc = (float*)d_out;
  float* o_cov = o_loc + (size_t)3 * n;
  dim3 grid((n + BLK - 1) / BLK), block(BLK);
  wexp_kernel<<<grid, block, 0, stream>>>(loc, cov1, vel, vcov, o_loc, o_cov, n);
}